// Linear_18494129177115
// MI455X (gfx1250) — compile-verified
//
#include <hip/hip_runtime.h>
#include <hip/hip_bf16.h>

typedef __attribute__((ext_vector_type(2))) float v2f;
typedef __attribute__((ext_vector_type(8))) float v8f;
typedef int v4i __attribute__((vector_size(16)));   // async builtin param type
typedef unsigned int u32x4 __attribute__((ext_vector_type(4)));
typedef int i32x8 __attribute__((ext_vector_type(8)));
typedef int i32x4 __attribute__((ext_vector_type(4)));

#define DD 4096      // inner dim D
#define OO 4096      // output dim O
#define MM 8192      // B*T tokens
#define EE 8         // experts
#define RR 16        // lora rank
#define ER 128       // E*R
#define SCALING_C 2.0f
#define LANG_BIAS_C 5.0f

#define ASG __attribute__((address_space(1)))
#define ASL __attribute__((address_space(3)))

#if defined(__AMDGCN__) && __has_builtin(__builtin_amdgcn_global_load_async_to_lds_b128)
#define HAS_ASYNC 1
#else
#define HAS_ASYNC 0
#endif

#if defined(__AMDGCN__) && __has_builtin(__builtin_amdgcn_tensor_load_to_lds)
#define HAS_TDM 1
#else
#define HAS_TDM 0
#endif

#if HAS_ASYNC
// async DMA of 16B/lane from global to LDS, tracked by ASYNCcnt
#define ASYNC_B128(gp, lp)                                                   \
  __builtin_amdgcn_global_load_async_to_lds_b128(                            \
      (ASG v4i*)(gp), (ASL v4i*)(lp), 0, 0)
#if __has_builtin(__builtin_amdgcn_s_wait_asynccnt)
#define WAIT_ASYNC() __builtin_amdgcn_s_wait_asynccnt(0)
#else
#define WAIT_ASYNC() asm volatile("s_wait_asynccnt 0x0" ::: "memory")
#endif
#endif

#if HAS_TDM
#if __has_builtin(__builtin_amdgcn_s_wait_tensorcnt)
#define WAIT_TENSOR() __builtin_amdgcn_s_wait_tensorcnt(0)
#else
#define WAIT_TENSOR() asm volatile("s_wait_tensorcnt 0x0" ::: "memory")
#endif

static __device__ __forceinline__ unsigned lds_offset(const void* p) {
  return (unsigned)(unsigned long long)(ASL const char*)p;
}

// TDM 2D tile load: tile_w x tile_h elements (f32) from a row-major tensor
// with row stride row_stride_elems, into LDS at lds_off with hardware row
// padding: after every 32 DWORDs stored, skip 4 DWORDs -> LDS stride 36 floats.
static __device__ __forceinline__ void tdm_load_tile_2d(
    const float* gaddr, unsigned lds_off, unsigned tile_w, unsigned tile_h,
    unsigned tensor_w, unsigned tensor_h, unsigned row_stride_elems)
{
  const unsigned long long ga = (unsigned long long)gaddr;
  u32x4 g0;
  g0[0] = 1u;                                    // count=1, user descriptor
  g0[1] = lds_off;                               // lds_addr (bytes)
  g0[2] = (unsigned)(ga & 0xffffffffu);          // global_addr[31:0]
  g0[3] = (unsigned)((ga >> 32) & 0x01ffffffu)   // global_addr[56:32]
        | (2u << 30);                            // type=2 ("image")
  i32x8 g1;
  g1[0] = (int)((2u << 16)                       // data_size = 4 bytes
              | (1u << 20)                       // pad_enable
              | (4u << 22)                       // pad_interval: 32 DWORDs
              | (3u << 25));                     // pad_amount:   4 DWORDs
  g1[1] = (int)((tensor_w & 0xffffu) << 16);     // tensor_dim0[15:0]
  g1[2] = (int)((tensor_w >> 16) | ((tensor_h & 0xffffu) << 16));
  g1[3] = (int)((tensor_h >> 16) | (tile_w << 16));   // tile_dim0
  g1[4] = (int)tile_h;                           // tile_dim1 (tile_dim2=0)
  g1[5] = (int)row_stride_elems;                 // tensor_dim0_stride[31:0]
  g1[6] = 0;
  g1[7] = 0;
  i32x4 z4 = {0, 0, 0, 0};                       // groups 2/3 unused (<=2D)
  i32x8 z8 = {0, 0, 0, 0, 0, 0, 0, 0};
  __builtin_amdgcn_tensor_load_to_lds(g0, g1, z4, z4, z8, 0);
}
#endif

static __device__ __forceinline__ v8f wmma4(v2f a, v2f b, v8f c) {
  // D = A(16x4 f32) * B(4x16 f32) + C(16x16 f32)
  return __builtin_amdgcn_wmma_f32_16x16x4_f32(
      /*neg_a=*/false, a, /*neg_b=*/false, b,
      /*c_mod=*/(short)0, c, /*reuse_a=*/false, /*reuse_b=*/false);
}

// ---------------------------------------------------------------------------
// Kernel 1: router. One wave32 per token. logits[e] = x[t]·Wr[e] (+ lang bias),
// top-2 + softmax -> dense gates[M, 8].
// ---------------------------------------------------------------------------
__global__ __launch_bounds__(256) void router_kernel(
    const float* __restrict__ X, const float* __restrict__ Wr,
    const int* __restrict__ lang, float* __restrict__ gates)
{
  const int t    = blockIdx.x * 8 + (threadIdx.x >> 5);
  const int lane = threadIdx.x & 31;

  float acc[EE];
  #pragma unroll
  for (int e = 0; e < EE; ++e) acc[e] = 0.0f;

  const float* xr = X + (size_t)t * DD;
  for (int d = lane * 4; d < DD; d += 128) {
    const float4 xv = *(const float4*)(xr + d);
    #pragma unroll
    for (int e = 0; e < EE; ++e) {
      const float4 wv = *(const float4*)(Wr + e * DD + d);
      acc[e] += xv.x * wv.x + xv.y * wv.y + xv.z * wv.z + xv.w * wv.w;
    }
  }
  #pragma unroll
  for (int e = 0; e < EE; ++e) {
    #pragma unroll
    for (int off = 16; off > 0; off >>= 1)
      acc[e] += __shfl_xor(acc[e], off, 32);
  }

  const int b   = t >> 11;          // t / 2048
  const int lid = lang[b];
  #pragma unroll
  for (int e = 0; e < EE; ++e)
    if (e == lid) acc[e] += LANG_BIAS_C;   // lid<0 -> no bias

  // top-2 (first index wins ties, matching lax.top_k)
  float l0 = -3.0e38f; int i0 = 0;
  #pragma unroll
  for (int e = 0; e < EE; ++e) if (acc[e] > l0) { l0 = acc[e]; i0 = e; }
  float l1 = -3.0e38f; int i1 = 0;
  #pragma unroll
  for (int e = 0; e < EE; ++e) if (e != i0 && acc[e] > l1) { l1 = acc[e]; i1 = e; }

  float w1 = __expf(l1 - l0);       // <= 1
  const float inv = 1.0f / (1.0f + w1);
  const float w0 = inv;
  w1 *= inv;

  if (lane < EE)
    gates[(size_t)t * EE + lane] = (lane == i0) ? w0 : ((lane == i1) ? w1 : 0.0f);
}

// ---------------------------------------------------------------------------
// Kernel 2: GH[t, k] = SCALING * gates[t, k/16] * (x[t]·A_flat[k])  (k = e*R+r)
// WMMA GEMM: block = 16 token rows, 8 waves, wave w owns k-columns [16w,16w+16)
// == expert e = w. X chunk (16x64) double-buffered in LDS via async-to-LDS;
// B-fragments for a whole chunk batch-preloaded into registers.
// ---------------------------------------------------------------------------
__global__ __launch_bounds__(256) void h_gemm_kernel(
    const float* __restrict__ X, const float* __restrict__ A,
    const float* __restrict__ gates, float* __restrict__ GH)
{
  __shared__ float sX[2][16 * 68];
  const int mblk = blockIdx.x * 16;
  const int tid  = threadIdx.x;
  const int lane = tid & 31, wv = tid >> 5;
  const int l15  = lane & 15, hi = lane >> 4;

  v8f acc = {};

  const float* Arow = A + (size_t)(wv * 16 + l15) * DD;  // B-operand column
  const float* Xg   = X + (size_t)mblk * DD;
  const int ldrow = tid >> 4;          // 0..15
  const int ldcol = (tid & 15) * 4;    // 0..60

#if HAS_ASYNC
  #define STAGE_H(bufi, kbase)                                               \
    ASYNC_B128(Xg + (size_t)ldrow * DD + (kbase) + ldcol,                    \
               &sX[bufi][ldrow * 68 + ldcol])

  int cur = 0;
  STAGE_H(0, 0);
  WAIT_ASYNC();
  __syncthreads();
  for (int k0 = 0; k0 < DD; k0 += 64) {
    if (k0 + 64 < DD) STAGE_H(cur ^ 1, k0 + 64);
    v2f bf[16];
    #pragma unroll
    for (int s = 0; s < 16; ++s)
      bf[s] = *(const v2f*)(Arow + k0 + s * 4 + 2 * hi);
    #pragma unroll
    for (int s = 0; s < 16; ++s) {
      v2f a = *(const v2f*)(&sX[cur][l15 * 68 + s * 4 + 2 * hi]);
      acc = wmma4(a, bf[s], acc);
    }
    WAIT_ASYNC();
    __syncthreads();
    cur ^= 1;
  }
#else
  for (int k0 = 0; k0 < DD; k0 += 64) {
    float4 xv = *(const float4*)(Xg + (size_t)ldrow * DD + k0 + ldcol);
    *(float4*)(&sX[0][ldrow * 68 + ldcol]) = xv;
    __syncthreads();
    v2f bf[16];
    #pragma unroll
    for (int s = 0; s < 16; ++s)
      bf[s] = *(const v2f*)(Arow + k0 + s * 4 + 2 * hi);
    #pragma unroll
    for (int s = 0; s < 16; ++s) {
      v2f a = *(const v2f*)(&sX[0][l15 * 68 + s * 4 + 2 * hi]);
      acc = wmma4(a, bf[s], acc);
    }
    __syncthreads();
  }
#endif

  // fold gate + scaling; wave index == expert index
  const int kcol = wv * 16 + l15;
  #pragma unroll
  for (int q = 0; q < 8; ++q) {
    const int t = mblk + q + 8 * hi;
    const float g = gates[(size_t)t * EE + wv];
    GH[(size_t)t * ER + kcol] = SCALING_C * g * acc[q];
  }
}

// ---------------------------------------------------------------------------
// Kernel 3: out = X @ W^T (+ fused GH @ Bc epilogue).
// WG tile 128(M) x 128(N); 8 waves as 4(M) x 2(N); wave computes 2x4 16x16
// tiles. X/W chunks (128 x 32) double-buffered in LDS; staged by the Tensor
// Data Mover (one descriptor per tile, HW pad -> stride-36 conflict-free
// layout) when available, else per-lane async-to-LDS.
// ---------------------------------------------------------------------------
__global__ __launch_bounds__(256) void main_gemm_kernel(
    const float* __restrict__ X, const float* __restrict__ W,
    const float* __restrict__ GH, const float* __restrict__ Bm,
    float* __restrict__ out)
{
  __shared__ float sX[2][128 * 36];
  __shared__ float sW[2][128 * 36];

  const int nblk = blockIdx.x * 128;
  const int mblk = blockIdx.y * 128;
  const int tid  = threadIdx.x;
  const int lane = tid & 31, wv = tid >> 5;
  const int l15  = lane & 15, hi = lane >> 4;
  const int wm   = (wv & 3) * 32;   // M offset of this wave inside WG tile
  const int wn   = (wv >> 2) * 64;  // N offset

  v8f acc[2][4] = {};

  const float* Xg = X + (size_t)mblk * DD;
  const float* Wg = W + (size_t)nblk * DD;

#if HAS_TDM
  // one TDM descriptor pair per K-chunk, issued by wave 0 only
  #define STAGE_MG(bufi, kbase)                                              \
    do {                                                                     \
      if (wv == 0) {                                                         \
        tdm_load_tile_2d(Xg + (kbase), lds_offset(&sX[bufi][0]),             \
                         32, 128, DD, MM, DD);                               \
        tdm_load_tile_2d(Wg + (kbase), lds_offset(&sW[bufi][0]),             \
                         32, 128, DD, OO, DD);                               \
      }                                                                      \
    } while (0)
  #define STAGE_WAIT()                                                       \
    do { if (wv == 0) WAIT_TENSOR(); __syncthreads(); } while (0)
#elif HAS_ASYNC
  const int ldr = tid >> 3;         // 0..31 (row within 32-row pass)
  const int ldc = (tid & 7) * 4;    // 0,4,...,28
  #define STAGE_MG(bufi, kbase)                                              \
    do {                                                                     \
      _Pragma("unroll")                                                      \
      for (int p = 0; p < 4; ++p) {                                          \
        const int row = p * 32 + ldr;                                        \
        ASYNC_B128(Xg + (size_t)row * DD + (kbase) + ldc,                    \
                   &sX[bufi][row * 36 + ldc]);                               \
        ASYNC_B128(Wg + (size_t)row * DD + (kbase) + ldc,                    \
                   &sW[bufi][row * 36 + ldc]);                               \
      }                                                                      \
    } while (0)
  #define STAGE_WAIT()                                                       \
    do { WAIT_ASYNC(); __syncthreads(); } while (0)
#else
  const int ldr = tid >> 3;
  const int ldc = (tid & 7) * 4;
  #define STAGE_MG(bufi, kbase)                                              \
    do {                                                                     \
      _Pragma("unroll")                                                      \
      for (int p = 0; p < 4; ++p) {                                          \
        const int row = p * 32 + ldr;                                        \
        float4 xv = *(const float4*)(Xg + (size_t)row * DD + (kbase) + ldc); \
        *(float4*)(&sX[bufi][row * 36 + ldc]) = xv;                          \
        float4 wvv = *(const float4*)(Wg + (size_t)row * DD + (kbase) + ldc);\
        *(float4*)(&sW[bufi][row * 36 + ldc]) = wvv;                         \
      }                                                                      \
    } while (0)
  #define STAGE_WAIT() __syncthreads()
#endif

  int cur = 0;
  STAGE_MG(0, 0);
  STAGE_WAIT();
  for (int k0 = 0; k0 < DD; k0 += 32) {
    if (k0 + 32 < DD) STAGE_MG(cur ^ 1, k0 + 32);
    #pragma unroll
    for (int kk = 0; kk < 32; kk += 4) {
      v2f a[2], b[4];
      #pragma unroll
      for (int i = 0; i < 2; ++i)
        a[i] = *(const v2f*)(&sX[cur][(wm + i * 16 + l15) * 36 + kk + 2 * hi]);
      #pragma unroll
      for (int j = 0; j < 4; ++j)
        b[j] = *(const v2f*)(&sW[cur][(wn + j * 16 + l15) * 36 + kk + 2 * hi]);
      #pragma unroll
      for (int i = 0; i < 2; ++i)
        #pragma unroll
        for (int j = 0; j < 4; ++j)
          acc[i][j] = wmma4(a[i], b[j], acc[i][j]);
    }
    STAGE_WAIT();
    cur ^= 1;
  }

  // fused LoRA epilogue: acc += GH[m, 0:128] @ Bc[0:128, o], Bc[k=e*16+r][o] = B[e,o,r]
  #pragma unroll
  for (int kk = 0; kk < ER; kk += 4) {
    const int kap = kk + 2 * hi;     // even, so float2 stays inside an r-chunk
    const int e = kap >> 4, r = kap & 15;
    v2f a[2], b[4];
    #pragma unroll
    for (int i = 0; i < 2; ++i) {
      const int mrow = mblk + wm + i * 16 + l15;
      a[i] = *(const v2f*)(GH + (size_t)mrow * ER + kap);
    }
    #pragma unroll
    for (int j = 0; j < 4; ++j) {
      const int o = nblk + wn + j * 16 + l15;
      b[j] = *(const v2f*)(Bm + ((size_t)e * OO + o) * RR + r);
    }
    #pragma unroll
    for (int i = 0; i < 2; ++i)
      #pragma unroll
      for (int j = 0; j < 4; ++j)
        acc[i][j] = wmma4(a[i], b[j], acc[i][j]);
  }

  // store (non-temporal: output is streamed once, keep L2 for X/W tiles)
  #pragma unroll
  for (int i = 0; i < 2; ++i) {
    #pragma unroll
    for (int j = 0; j < 4; ++j) {
      const int o = nblk + wn + j * 16 + l15;
      #pragma unroll
      for (int q = 0; q < 8; ++q) {
        const int mrow = mblk + wm + i * 16 + q + 8 * hi;
        __builtin_nontemporal_store(acc[i][j][q], &out[(size_t)mrow * OO + o]);
      }
    }
  }
}

// ---------------------------------------------------------------------------
extern "C" void kernel_launch(void* const* d_in, const int* in_sizes, int n_in,
                              void* d_out, int out_size, void* d_ws, size_t ws_size,
                              hipStream_t stream) {
  (void)in_sizes; (void)n_in; (void)out_size; (void)ws_size;
  const float* X    = (const float*)d_in[0];   // [4,2048,4096]
  const int*   lang = (const int*)d_in[1];     // [4]
  const float* W    = (const float*)d_in[2];   // [4096,4096]
  const float* Wr   = (const float*)d_in[3];   // [8,4096]
  const float* A    = (const float*)d_in[4];   // [8,16,4096] == [128,4096]
  const float* Bm   = (const float*)d_in[5];   // [8,4096,16]
  float* out = (float*)d_out;                  // [4,2048,4096]

  float* gates = (float*)d_ws;                 // MM*EE  = 64K floats (256 KB)
  float* GH    = gates + (size_t)MM * EE;      // MM*ER  = 1M floats  (4 MB)

  router_kernel<<<MM / 8, 256, 0, stream>>>(X, Wr, lang, gates);
  h_gemm_kernel<<<MM / 16, 256, 0, stream>>>(X, A, gates, GH);
  dim3 grid(OO / 128, MM / 128);
  main_gemm_kernel<<<grid, 256, 0, stream>>>(X, W, GH, Bm, out);
}